// EdgeQLayer_29008209117743
// MI455X (gfx1250) — compile-verified
//
#include <hip/hip_runtime.h>
#include <hip/hip_bf16.h>

#define NEG_SLOPE 0.01f

typedef float v2f __attribute__((ext_vector_type(2)));
typedef float v8f __attribute__((ext_vector_type(8)));

// ---------------------------------------------------------------------------
// Phase 1a: per-chunk partial sums of embeddings over the node axis.
// grid = B * nch blocks, 128 threads (one per d). Deterministic (no atomics).
// ---------------------------------------------------------------------------
__global__ void gsum_partial(const float* __restrict__ emb, float* __restrict__ part,
                             int N, int nch, int chunk) {
    const int bid = blockIdx.x;
    const int b   = bid / nch;
    const int c   = bid % nch;
    const int t   = threadIdx.x;           // d in [0,128)
    const int n0  = c * chunk;
    const int n1  = (n0 + chunk < N) ? (n0 + chunk) : N;
    const float* p = emb + ((size_t)b * N + n0) * 128 + t;
    float acc = 0.f;
    for (int n = n0; n < n1; ++n) { acc += *p; p += 128; }
    part[(size_t)bid * 128 + t] = acc;
}

// Phase 1b: fold partials into g[b][d]. grid = B, 128 threads.
__global__ void gsum_final(const float* __restrict__ part, float* __restrict__ g, int nch) {
    const int b = blockIdx.x, t = threadIdx.x;
    float acc = 0.f;
    for (int c = 0; c < nch; ++c) acc += part[((size_t)b * nch + c) * 128 + t];
    g[b * 128 + t] = acc;
}

// ---------------------------------------------------------------------------
// Phase 2: head. g_proj = g@W6; qg[b] = lrelu(g_proj)·w5_g; noop = g·w_noop.
// 1 block, 128 threads. Writes qg to ws and noop_q straight into d_out.
// ---------------------------------------------------------------------------
__global__ void head_kernel(const float* __restrict__ g, const float* __restrict__ W6,
                            const float* __restrict__ w5, const float* __restrict__ w_noop,
                            float* __restrict__ qg, float* __restrict__ out, int E) {
    __shared__ float red[128];
    const int t = threadIdx.x;
    for (int b = 0; b < 4; ++b) {
        // column t of g_proj for batch b
        float acc = 0.f;
        for (int d = 0; d < 128; ++d) acc += g[b * 128 + d] * W6[d * 128 + t];
        float lr = (acc >= 0.f) ? acc : NEG_SLOPE * acc;
        red[t] = lr * w5[t];               // w5_g = w5[0:128]
        __syncthreads();
        for (int s = 64; s > 0; s >>= 1) { if (t < s) red[t] += red[t + s]; __syncthreads(); }
        if (t == 0) qg[b] = red[0];
        __syncthreads();
        red[t] = g[b * 128 + t] * w_noop[t];
        __syncthreads();
        for (int s = 64; s > 0; s >>= 1) { if (t < s) red[t] += red[t + s]; __syncthreads(); }
        if (t == 0) out[(size_t)b * (E + 1) + E] = red[0];   // noop_q column
        __syncthreads();
    }
}

// ---------------------------------------------------------------------------
// Phase 3: node_proj = emb @ W7 via V_WMMA_F32_16X16X4_F32, fused epilogue:
// qi = lrelu(node_proj)·w5_i, qj = lrelu(node_proj)·w5_j.
// 256 threads = 8 waves; each wave owns 16 rows x 128 cols (8 accum tiles).
// Dynamic LDS layout (floats):
//   [0,18432)        W7 padded to row stride 144 (bank-conflict-free B reads)
//   [18432,18560)    w5_i
//   [18560,18688)    w5_j
//   [18688, +8*2112) per-wave A tile, 16 rows x stride 132 (conflict-free)
// ---------------------------------------------------------------------------
__global__ void __launch_bounds__(256)
node_proj_kernel(const float* __restrict__ emb, const float* __restrict__ W7,
                 const float* __restrict__ w5, float* __restrict__ qi,
                 float* __restrict__ qj, int rowsTotal) {
    extern __shared__ float smem[];
    float* W7p = smem;                      // 128 x 144
    float* w5i = smem + 18432;
    float* w5j = smem + 18560;
    float* Aw0 = smem + 18688;

    const int tid = threadIdx.x;

    // Cooperative W7 load: 16384 floats as 4096 float4, coalesced.
    for (int i4 = tid; i4 < 4096; i4 += 256) {
        const int r = i4 >> 5;              // 32 float4 per source row
        const int c = (i4 & 31) << 2;
        const float4 vv = ((const float4*)W7)[i4];
        float* dst = &W7p[r * 144 + c];
        dst[0] = vv.x; dst[1] = vv.y; dst[2] = vv.z; dst[3] = vv.w;
    }
    if (tid < 128) { w5i[tid] = w5[128 + tid]; w5j[tid] = w5[256 + tid]; }
    __syncthreads();

    const int wave = tid >> 5;
    const int lane = tid & 31;
    const int rowbase = blockIdx.x * 128 + wave * 16;
    if (rowbase >= rowsTotal) return;       // wave-uniform: EXEC stays full for WMMA

    // Stage this wave's 16 A rows into LDS (coalesced float4 loads).
    float* Aw = Aw0 + wave * 2112;          // 16 x 132
    {
        const float4* src = (const float4*)(emb + (size_t)rowbase * 128);
        for (int j = 0; j < 16; ++j) {
            const float4 vv = src[j * 32 + lane];
            float* dst = &Aw[j * 132 + lane * 4];
            dst[0] = vv.x; dst[1] = vv.y; dst[2] = vv.z; dst[3] = vv.w;
        }
    }

    const int half = lane >> 4;             // 0: K+0/K+1 ; 1: K+2/K+3 (ISA A/B layout)
    const int mrow = lane & 15;

    float wi[8], wj[8];
#pragma unroll
    for (int ct = 0; ct < 8; ++ct) { wi[ct] = w5i[ct * 16 + mrow]; wj[ct] = w5j[ct * 16 + mrow]; }

    v8f acc[8];
#pragma unroll
    for (int ct = 0; ct < 8; ++ct)
#pragma unroll
        for (int v = 0; v < 8; ++v) acc[ct][v] = 0.f;

    // K loop: 32 steps of K=4. A = 16x4 (2 VGPRs), B = 4x16 (2 VGPRs).
    for (int kt = 0; kt < 32; ++kt) {
        v2f a;
        const int ab = mrow * 132 + kt * 4 + half * 2;
        a.x = Aw[ab];                       // A[M=mrow][K0 or K2]
        a.y = Aw[ab + 1];                   // A[M=mrow][K1 or K3]
        const int bb = (kt * 4 + half * 2) * 144 + mrow;
#pragma unroll
        for (int ct = 0; ct < 8; ++ct) {
            v2f bv;
            bv.x = W7p[bb + ct * 16];       // B[K0/K2][N=mrow of tile ct]
            bv.y = W7p[bb + 144 + ct * 16]; // B[K1/K3][N=mrow]
            acc[ct] = __builtin_amdgcn_wmma_f32_16x16x4_f32(
                false, a, false, bv, (short)0, acc[ct], false, false);
        }
    }

    // Fused epilogue: leaky-ReLU, dot with w5_i/w5_j, 16-lane reductions.
    // acc[ct][v] at lane L = node_proj[row = rowbase + v + 8*half][col = ct*16 + mrow]
#pragma unroll
    for (int v = 0; v < 8; ++v) {
        float si = 0.f, sj = 0.f;
#pragma unroll
        for (int ct = 0; ct < 8; ++ct) {
            const float x = acc[ct][v];
            const float t = (x >= 0.f) ? x : NEG_SLOPE * x;
            si += t * wi[ct];
            sj += t * wj[ct];
        }
        for (int m = 8; m >= 1; m >>= 1) {  // reduce within each 16-lane half
            si += __shfl_xor(si, m, 32);
            sj += __shfl_xor(sj, m, 32);
        }
        if (mrow == 0) {
            const int row = rowbase + v + half * 8;
            qi[row] = si;
            qj[row] = sj;
        }
    }
}

// ---------------------------------------------------------------------------
// Phase 4: edge gather. out[b][e] = qg[b] + qi[b,u] + qj[b,v]. L2-resident.
// ---------------------------------------------------------------------------
__global__ void edge_kernel(const int* __restrict__ edges, const float* __restrict__ qg,
                            const float* __restrict__ qi, const float* __restrict__ qj,
                            float* __restrict__ out, int E, int N, int total) {
    const int t = blockIdx.x * blockDim.x + threadIdx.x;
    if (t >= total) return;
    const int b = t / E;
    const int e = t - b * E;
    const int u = edges[2 * e];
    const int v = edges[2 * e + 1];
    out[(size_t)b * (E + 1) + e] = qg[b] + qi[(size_t)b * N + u] + qj[(size_t)b * N + v];
}

extern "C" void kernel_launch(void* const* d_in, const int* in_sizes, int n_in,
                              void* d_out, int out_size, void* d_ws, size_t ws_size,
                              hipStream_t stream) {
    const float* emb    = (const float*)d_in[0];
    const int*   edges  = (const int*)d_in[1];
    const float* W6     = (const float*)d_in[2];
    const float* W7     = (const float*)d_in[3];
    const float* w5     = (const float*)d_in[4];
    const float* w_noop = (const float*)d_in[5];
    float* out = (float*)d_out;

    const int B = 4, D = 128;
    const int N = in_sizes[0] / (B * D);    // 20000
    const int E = in_sizes[1] / 2;          // 100000

    // Workspace layout (floats); everything written before read each call.
    float* ws   = (float*)d_ws;
    const int CHUNK = 500;
    const int nch   = (N + CHUNK - 1) / CHUNK;
    float* g    = ws;                       // B*128
    float* qg   = ws + 512;                 // B (padded)
    float* part = ws + 640;                 // B*nch*128
    float* qi   = part + (size_t)B * nch * 128;  // B*N
    float* qj   = qi + (size_t)B * N;            // B*N

    gsum_partial<<<B * nch, 128, 0, stream>>>(emb, part, N, nch, CHUNK);
    gsum_final<<<B, 128, 0, stream>>>(part, g, nch);
    head_kernel<<<1, 128, 0, stream>>>(g, W6, w5, w_noop, qg, out, E);

    const int rowsTotal = B * N;            // 80000
    const int blocks    = (rowsTotal + 127) / 128;   // 625
    const size_t shmem  = (size_t)(128 * 144 + 128 + 128 + 8 * 16 * 132) * sizeof(float);
    node_proj_kernel<<<blocks, 256, shmem, stream>>>(emb, W7, w5, qi, qj, rowsTotal);

    const int total = B * E;
    edge_kernel<<<(total + 255) / 256, 256, 0, stream>>>(edges, qg, qi, qj, out, E, N, total);
}